// TextureMapper_9895604650546
// MI455X (gfx1250) — compile-verified
//
#include <hip/hip_runtime.h>

// ---------------------------------------------------------------------------
// TextureMapper: 4-level bilinear gather (C=16) + SH modulation, NCHW output.
// Memory-bound gather kernel; CDNA5 paths used: TDM tensor_load_to_lds for the
// streamed uv tiles (per-wave 256B DMA into LDS, overlapped with SH loads),
// s_wait_tensorcnt, ds_load for uv, global_load_b128 gathers.
// L2 policy: textures cached (RT); streaming sh loads and output stores are
// non-temporal so the ~89MB of gathered texture data stays resident in the
// 192MB L2 while ~105MB of stream traffic passes through.
// ---------------------------------------------------------------------------

typedef unsigned int u32x4 __attribute__((ext_vector_type(4)));
typedef int          i32x8 __attribute__((ext_vector_type(8)));
typedef int          i32x4 __attribute__((ext_vector_type(4)));

#define HW_BITS 18            // 512*512 pixels per image
#define HW_PIX  (1 << HW_BITS)

// Per-level bilinear setup: 4 element offsets (x16 channels) + 4 weights.
// Matches the reference exactly: clip, edge fixup (x0 - (x0==x1)), valid mask.
__device__ __forceinline__ void level_setup(float u, float v, const int S,
                                            unsigned off[4], float w[4]) {
  const float scale = (float)(S - 1);
  const float sx = u * scale;
  const float sy = scale - v * scale;     // flip v axis
  const float valid =
      ((sx >= 0.0f) && (sx <= scale) && (sy >= 0.0f) && (sy <= scale)) ? 1.0f : 0.0f;
  int x0 = (int)floorf(sx);
  int y0 = (int)floorf(sy);
  int x1 = x0 + 1, y1 = y0 + 1;
  x0 = min(max(x0, 0), S - 1);
  x1 = min(max(x1, 0), S - 1);
  y0 = min(max(y0, 0), S - 1);
  y1 = min(max(y1, 0), S - 1);
  const float x0f = (float)(x0 - (x0 == x1 ? 1 : 0));
  const float y0f = (float)(y0 - (y0 == y1 ? 1 : 0));
  const float wx1 = (float)x1 - sx;
  const float wx0 = sx - x0f;
  const float wy1 = (float)y1 - sy;
  const float wy0 = sy - y0f;
  w[0] = wx1 * wy1 * valid;   // I00 @ (y0,x0)
  w[1] = wx1 * wy0 * valid;   // I10 @ (y1,x0)
  w[2] = wx0 * wy1 * valid;   // I01 @ (y0,x1)
  w[3] = wx0 * wy0 * valid;   // I11 @ (y1,x1)
  off[0] = (unsigned)((y0 * S + x0) << 4);  // *16 channels
  off[1] = (unsigned)((y1 * S + x0) << 4);
  off[2] = (unsigned)((y0 * S + x1) << 4);
  off[3] = (unsigned)((y1 * S + x1) << 4);
}

// Gather 4 taps (one channel-group of 4 floats each, 16B-aligned b128 loads)
// and FMA into the accumulator. Default (RT) policy: keep textures in L2.
__device__ __forceinline__ void accum4(const float* __restrict__ tex,
                                       const unsigned off[4], const float w[4],
                                       int cg, float4& a) {
#pragma unroll
  for (int t = 0; t < 4; ++t) {
    const float4 d = *(const float4*)(tex + off[t] + (cg << 2));
    a.x = fmaf(w[t], d.x, a.x);
    a.y = fmaf(w[t], d.y, a.y);
    a.z = fmaf(w[t], d.z, a.z);
    a.w = fmaf(w[t], d.w, a.w);
  }
}

// Issue a TDM DMA: copy `td0`-bounded 64-float 1-D tile from global `gaddr`
// into LDS byte offset `lds_addr`. D# bit layout per CDNA5 ISA §8.3/§8.4.
__device__ __forceinline__ void tdm_load_row_to_lds(unsigned long long gaddr,
                                                    unsigned lds_addr,
                                                    unsigned td0) {
  u32x4 g0;
  g0[0] = 1u;                                        // count=1, user mode
  g0[1] = lds_addr;                                  // lds_addr [63:32]
  g0[2] = (unsigned)(gaddr & 0xFFFFFFFFull);         // global_addr lo
  g0[3] = (unsigned)((gaddr >> 32) & 0x1FFFFFFull)   // global_addr hi [120:96]
        | (2u << 30);                                // type=2 ("image") [127:126]
  i32x8 g1;
  g1[0] = (int)(2u << 16);                           // data_size=2 (4 bytes)
  g1[1] = (int)((td0 & 0xFFFFu) << 16);              // tensor_dim0[15:0] @ [63:48]
  g1[2] = (int)((td0 >> 16) | (1u << 16));           // tensor_dim0[31:16] | tensor_dim1=1
  g1[3] = (int)(64u << 16);                          // tile_dim0=64 @ [127:112]
  g1[4] = 1;                                         // tile_dim1=1
  g1[5] = 64;                                        // tensor_dim0_stride lo
  g1[6] = 0;                                         // stride hi | dim1_stride lo
  g1[7] = 0;
  const i32x4 z4 = {0, 0, 0, 0};                     // groups 2/3 unused (<=2D)
  const i32x8 z8 = {0, 0, 0, 0, 0, 0, 0, 0};
  // clang-23 / amdgpu-toolchain form: 6 args (extra int32x8 before cpol).
  __builtin_amdgcn_tensor_load_to_lds(g0, g1, z4, z4, z8, 0);
}

__global__ __launch_bounds__(256) void texmap_sh_kernel(
    const float* __restrict__ uv,   // [N,H,W,2]
    const float* __restrict__ sh,   // [N,H,W,9]
    const float* __restrict__ t0,   // [1024,1024,16]
    const float* __restrict__ t1,   // [512,512,16]
    const float* __restrict__ t2,   // [256,256,16]
    const float* __restrict__ t3,   // [128,128,16]
    float* __restrict__ out,        // [N,16,H,W]
    int npix) {
  __shared__ __align__(16) float lds_uv[512];        // 256 px * 2 floats
  const int tid  = threadIdx.x;
  const int pix0 = (int)(blockIdx.x << 8);
  const int wave = __builtin_amdgcn_readfirstlane(tid >> 5);

  // --- TDM: each wave DMAs its own 32 pixels of uv (256B) into its LDS slice.
  {
    const unsigned basef = (unsigned)((pix0 + wave * 32) * 2);
    const unsigned totf  = (unsigned)npix * 2u;
    const unsigned td0   = (totf > basef) ? (totf - basef) : 0u;  // OOB -> zeros
    const unsigned long long ga =
        (unsigned long long)(uintptr_t)(uv + (size_t)basef);
    const unsigned lb =
        (unsigned)(uintptr_t)(&lds_uv[0]) + (unsigned)(wave * 256);
    tdm_load_row_to_lds(ga, lb, td0);
  }

  const int p = pix0 + tid;

  // --- Overlap with the DMA: pull this lane's 9 SH floats (read-once stream,
  //     non-temporal so it does not displace texture lines in L2).
  float s9[9];
  if (p < npix) {
    const float* shp = sh + (size_t)p * 9u;
#pragma unroll
    for (int i = 0; i < 9; ++i) s9[i] = __builtin_nontemporal_load(&shp[i]);
  }

  // --- Wait for the tensor DMA, then read uv from LDS (ds_load_b64).
  __builtin_amdgcn_s_wait_tensorcnt(0);
  const float2 uvv = *(const float2*)&lds_uv[2 * tid];

  if (p < npix) {
    const float u = uvv.x;
    const float v = uvv.y;

    // Per-level tap offsets + weights held in VGPRs.
    unsigned o0[4], o1[4], o2[4], o3[4];
    float    w0[4], w1[4], w2[4], w3[4];
    level_setup(u, v, 1024, o0, w0);
    level_setup(u, v, 512,  o1, w1);
    level_setup(u, v, 256,  o2, w2);
    level_setup(u, v, 128,  o3, w3);

    float acc[16];
#pragma unroll
    for (int cg = 0; cg < 4; ++cg) {
      float4 a = make_float4(0.0f, 0.0f, 0.0f, 0.0f);
      accum4(t0, o0, w0, cg, a);   // 16 global_load_b128 gathers per cg,
      accum4(t1, o1, w1, cg, a);   // all in flight before the FMAs retire
      accum4(t2, o2, w2, cg, a);
      accum4(t3, o3, w3, cg, a);
      acc[cg * 4 + 0] = a.x;
      acc[cg * 4 + 1] = a.y;
      acc[cg * 4 + 2] = a.z;
      acc[cg * 4 + 3] = a.w;
    }

    // SH modulation of channels 3..11.
#pragma unroll
    for (int i = 0; i < 9; ++i) acc[3 + i] *= s9[i];

    // Coalesced per-channel-plane stores: out[n][c][h][w]. Write-once output:
    // non-temporal stores keep L2 capacity for the gathered textures.
    const int n       = p >> HW_BITS;
    const unsigned hw = (unsigned)p & (HW_PIX - 1);
    float* op = out + ((size_t)n << (HW_BITS + 4)) + hw;
#pragma unroll
    for (int c = 0; c < 16; ++c)
      __builtin_nontemporal_store(acc[c], &op[((size_t)c) << HW_BITS]);
  }
}

extern "C" void kernel_launch(void* const* d_in, const int* in_sizes, int n_in,
                              void* d_out, int out_size, void* d_ws, size_t ws_size,
                              hipStream_t stream) {
  (void)n_in; (void)out_size; (void)d_ws; (void)ws_size;
  const float* uv = (const float*)d_in[0];
  const float* sh = (const float*)d_in[1];
  const float* t0 = (const float*)d_in[2];
  const float* t1 = (const float*)d_in[3];
  const float* t2 = (const float*)d_in[4];
  const float* t3 = (const float*)d_in[5];
  float* out = (float*)d_out;

  const int npix   = in_sizes[0] / 2;           // N*H*W = 1,048,576
  const int blocks = (npix + 255) / 256;        // 4096 full blocks

  hipLaunchKernelGGL(texmap_sh_kernel, dim3(blocks), dim3(256), 0, stream,
                     uv, sh, t0, t1, t2, t3, out, npix);
}